// BahdanauAttnDecoderRNN_70798240907338
// MI455X (gfx1250) — compile-verified
//
#include <hip/hip_runtime.h>
#include <math.h>

#define S_LEN 2048
#define B_SZ 64
#define H_SZ 1024
#define DIN_SZ 1024
#define K_IH (DIN_SZ + H_SZ)    // 2048
#define N3H (3 * H_SZ)          // 3072
#define NCHUNK 16
#define SCHUNK (S_LEN / NCHUNK) // 128
#define SPI 4                   // s-rows processed per barrier

typedef __attribute__((ext_vector_type(2))) float v2f;
typedef __attribute__((ext_vector_type(8))) float v8f;

// ---------------------------------------------------------------------------
// Kernel A: streaming scores + online softmax + unnormalized context partials.
// One WG per (batch, S-chunk); 256 threads = 8 wave32; each thread owns 4
// contiguous H elements. encoder_outputs is read exactly once (512 MB total).
// 4 s-rows per iteration with a double-buffered LDS reduction: one barrier
// per 4 rows, 4 float4 row fragments in flight per thread.
// ---------------------------------------------------------------------------
__global__ __launch_bounds__(256) void attn_partial_kernel(
    const float* __restrict__ h,        // (B, H) = last_hidden[0]
    const float* __restrict__ enc,      // (S, B, H)
    float* __restrict__ scores,         // (B, S) raw scores
    float* __restrict__ ctx_part,       // (B*NCHUNK, H) unnormalized
    float* __restrict__ ml_part)        // (B*NCHUNK, 2) {m, l}
{
    const int tid   = threadIdx.x;
    const int b     = blockIdx.x & (B_SZ - 1);
    const int chunk = blockIdx.x >> 6;
    const int s0    = chunk * SCHUNK;
    const int lane  = tid & 31;
    const int wave  = tid >> 5;

    __shared__ float red[2][8][SPI];

    const float4 hv = *(const float4*)(h + b * H_SZ + tid * 4);

    float4 c = make_float4(0.f, 0.f, 0.f, 0.f);
    float m = -INFINITY;
    float l = 0.f;

    int buf = 0;
    for (int s = s0; s < s0 + SCHUNK; s += SPI, buf ^= 1) {
        float4 e[SPI];
        float  p[SPI];

#pragma unroll
        for (int j = 0; j < SPI; ++j) {
            const float* row =
                enc + ((size_t)(s + j) * B_SZ + b) * H_SZ + tid * 4;
            // speculative prefetch 2 iterations (8 rows) ahead, same column
            __builtin_prefetch(row + (size_t)(2 * SPI) * B_SZ * H_SZ, 0, 3);
            e[j] = *(const float4*)row;
        }

#pragma unroll
        for (int j = 0; j < SPI; ++j) {
            float pp = hv.x * e[j].x + hv.y * e[j].y +
                       hv.z * e[j].z + hv.w * e[j].w;
            pp += __shfl_xor(pp, 16, 32);
            pp += __shfl_xor(pp,  8, 32);
            pp += __shfl_xor(pp,  4, 32);
            pp += __shfl_xor(pp,  2, 32);
            pp += __shfl_xor(pp,  1, 32);
            p[j] = pp;
        }
        if (lane == 0) {
#pragma unroll
            for (int j = 0; j < SPI; ++j) red[buf][wave][j] = p[j];
        }
        __syncthreads();

        float score[SPI];
#pragma unroll
        for (int j = 0; j < SPI; ++j) {
            float sc = 0.f;
#pragma unroll
            for (int w = 0; w < 8; ++w) sc += red[buf][w][j];  // LDS broadcast
            score[j] = sc;
        }
        if (tid == 0) {
#pragma unroll
            for (int j = 0; j < SPI; ++j)
                scores[b * S_LEN + s + j] = score[j];
        }

        // online softmax updates (uniform across WG)
#pragma unroll
        for (int j = 0; j < SPI; ++j) {
            if (score[j] > m) {
                const float sc = __expf(m - score[j]); // exp(-inf)=0 first time
                c.x *= sc; c.y *= sc; c.z *= sc; c.w *= sc;
                l *= sc;
                m = score[j];
            }
            const float pe = __expf(score[j] - m);
            l += pe;
            c.x += pe * e[j].x; c.y += pe * e[j].y;
            c.z += pe * e[j].z; c.w += pe * e[j].w;
        }
        // double-buffered red[] needs no trailing barrier: next write to this
        // buffer happens after the next iteration's barrier.
    }

    const int pidx = b * NCHUNK + chunk;
    *(float4*)(ctx_part + (size_t)pidx * H_SZ + tid * 4) = c;
    if (tid == 0) {
        ml_part[pidx * 2 + 0] = m;
        ml_part[pidx * 2 + 1] = l;
    }
}

// ---------------------------------------------------------------------------
// Kernel B: merge per-chunk partials -> normalized context + final (m, l).
// One WG per batch.
// ---------------------------------------------------------------------------
__global__ __launch_bounds__(256) void attn_combine_kernel(
    const float* __restrict__ ctx_part,
    const float* __restrict__ ml_part,
    float* __restrict__ context,        // (B, H)
    float* __restrict__ ml_fin)         // (B, 2)
{
    const int b   = blockIdx.x;
    const int tid = threadIdx.x;

    float m = -INFINITY;
    for (int i = 0; i < NCHUNK; ++i)
        m = fmaxf(m, ml_part[(b * NCHUNK + i) * 2]);

    float l = 0.f;
    float4 c = make_float4(0.f, 0.f, 0.f, 0.f);
    for (int i = 0; i < NCHUNK; ++i) {
        const float mi = ml_part[(b * NCHUNK + i) * 2 + 0];
        const float li = ml_part[(b * NCHUNK + i) * 2 + 1];
        const float sc = __expf(mi - m);
        l += li * sc;
        const float4 ci =
            *(const float4*)(ctx_part + (size_t)(b * NCHUNK + i) * H_SZ + tid * 4);
        c.x += sc * ci.x; c.y += sc * ci.y; c.z += sc * ci.z; c.w += sc * ci.w;
    }
    const float inv = 1.f / l;
    c.x *= inv; c.y *= inv; c.z *= inv; c.w *= inv;
    *(float4*)(context + (size_t)b * H_SZ + tid * 4) = c;
    if (tid == 0) {
        ml_fin[b * 2 + 0] = m;
        ml_fin[b * 2 + 1] = l;
    }
}

// ---------------------------------------------------------------------------
// Kernel C: attn_weights[b,0,s] = exp(score - m) / l
// ---------------------------------------------------------------------------
__global__ __launch_bounds__(256) void attn_weights_kernel(
    const float* __restrict__ scores,
    const float* __restrict__ ml_fin,
    float* __restrict__ out_attn)       // (B, 1, S) flat
{
    const int idx = blockIdx.x * 256 + threadIdx.x; // b*S + s
    const int b = idx >> 11;
    const float m = ml_fin[b * 2 + 0];
    const float l = ml_fin[b * 2 + 1];
    out_attn[idx] = __expf(scores[idx] - m) / l;
}

// ---------------------------------------------------------------------------
// Kernel D: gi = [x|context] @ W_ih^T + b_ih ; gh = h @ W_hh^T + b_hh
// via V_WMMA_F32_16X16X4_F32 (full fp32). Each wave owns one 16x16 C tile:
// 4 M-tiles (batch) x 2 N-tiles per 256-thread WG; grid.x = 3072/32 = 96.
// A fragment (16x4 f32, 2 VGPRs): lane&15 = M row, lane>>4 selects K-pair.
// B fragment (4x16 f32, 2 VGPRs): lane&15 = N col, lane>>4 selects K-pair.
// The concat [x | context] is handled as two separate K-loops so the inner
// loop has no per-iteration select.
// ---------------------------------------------------------------------------
__global__ __launch_bounds__(256) void gru_gemm_kernel(
    const float* __restrict__ x,        // (B, DIN)
    const float* __restrict__ context,  // (B, H)
    const float* __restrict__ h,        // (B, H)
    const float* __restrict__ W_ih,     // (3H, DIN+H)
    const float* __restrict__ W_hh,     // (3H, H)
    const float* __restrict__ b_ih,     // (3H)
    const float* __restrict__ b_hh,     // (3H)
    float* __restrict__ gi,             // (B, 3H)
    float* __restrict__ gh)             // (B, 3H)
{
    const int tid   = threadIdx.x;
    const int lane  = tid & 31;
    const int wave  = tid >> 5;
    const int mtile = wave & 3;
    const int nt    = wave >> 2;
    const int m0    = mtile * 16;
    const int n0    = blockIdx.x * 32 + nt * 16;

    const int r15  = lane & 15;
    const int half = lane >> 4;
    const int arow = m0 + r15;   // batch row feeding the A fragment
    const int ncol = n0 + r15;   // weight row feeding the B fragment
    const int koff = 2 * half;   // this lane's K-pair offset (even -> aligned)

    // Hoisted per-lane base pointers
    const float* ax = x       + (size_t)arow * DIN_SZ + koff;
    const float* ac = context + (size_t)arow * H_SZ   + koff;
    const float* ah = h       + (size_t)arow * H_SZ   + koff;
    const float* wi = W_ih    + (size_t)ncol * K_IH   + koff;
    const float* wh = W_hh    + (size_t)ncol * H_SZ   + koff;

    // ---- gi: K = 2048 over concat [x | context] --------------------------
    v8f acc = {0.f, 0.f, 0.f, 0.f, 0.f, 0.f, 0.f, 0.f};
#pragma unroll 8
    for (int k = 0; k < DIN_SZ; k += 4) {
        const v2f a  = *(const v2f*)(ax + k);
        const v2f bb = *(const v2f*)(wi + k);
        acc = __builtin_amdgcn_wmma_f32_16x16x4_f32(
            false, a, false, bb, (short)0, acc, false, false);
    }
#pragma unroll 8
    for (int k = 0; k < H_SZ; k += 4) {
        const v2f a  = *(const v2f*)(ac + k);
        const v2f bb = *(const v2f*)(wi + DIN_SZ + k);
        acc = __builtin_amdgcn_wmma_f32_16x16x4_f32(
            false, a, false, bb, (short)0, acc, false, false);
    }
    {
        const int nn = n0 + r15;
        const float bias = b_ih[nn];
#pragma unroll
        for (int r = 0; r < 8; ++r) {
            const int mm = m0 + r + 8 * half;
            gi[(size_t)mm * N3H + nn] = acc[r] + bias;
        }
    }

    // ---- gh: K = 1024 over h --------------------------------------------
    v8f acc2 = {0.f, 0.f, 0.f, 0.f, 0.f, 0.f, 0.f, 0.f};
#pragma unroll 8
    for (int k = 0; k < H_SZ; k += 4) {
        const v2f a  = *(const v2f*)(ah + k);
        const v2f bb = *(const v2f*)(wh + k);
        acc2 = __builtin_amdgcn_wmma_f32_16x16x4_f32(
            false, a, false, bb, (short)0, acc2, false, false);
    }
    {
        const int nn = n0 + r15;
        const float bias = b_hh[nn];
#pragma unroll
        for (int r = 0; r < 8; ++r) {
            const int mm = m0 + r + 8 * half;
            gh[(size_t)mm * N3H + nn] = acc2[r] + bias;
        }
    }
}

// ---------------------------------------------------------------------------
// Kernel E: GRU gating -> output (slot 0) and hidden (slot 1)
// ---------------------------------------------------------------------------
__global__ __launch_bounds__(256) void gru_gate_kernel(
    const float* __restrict__ gi,
    const float* __restrict__ gh,
    const float* __restrict__ h,
    float* __restrict__ out)            // d_out base
{
    const int idx = blockIdx.x * 256 + threadIdx.x; // b*H + j
    const int b = idx >> 10;
    const int j = idx & (H_SZ - 1);
    const float* gib = gi + (size_t)b * N3H;
    const float* ghb = gh + (size_t)b * N3H;

    const float ir  = gib[j];
    const float iz  = gib[H_SZ + j];
    const float in_ = gib[2 * H_SZ + j];
    const float hr  = ghb[j];
    const float hz  = ghb[H_SZ + j];
    const float hn  = ghb[2 * H_SZ + j];

    const float r = 1.f / (1.f + __expf(-(ir + hr)));
    const float z = 1.f / (1.f + __expf(-(iz + hz)));
    const float n = tanhf(in_ + r * hn);
    const float hnew = (1.f - z) * n + z * h[idx];

    out[idx] = hnew;                    // output (B,H)
    out[B_SZ * H_SZ + idx] = hnew;      // hidden (1,B,H)
}

// ---------------------------------------------------------------------------
extern "C" void kernel_launch(void* const* d_in, const int* in_sizes, int n_in,
                              void* d_out, int out_size, void* d_ws, size_t ws_size,
                              hipStream_t stream) {
    const float* x    = (const float*)d_in[0];   // (1,B,DIN)
    const float* lh   = (const float*)d_in[1];   // (1,B,H)
    const float* enc  = (const float*)d_in[2];   // (S,B,H)
    const float* W_ih = (const float*)d_in[3];   // (3H, DIN+H)
    const float* W_hh = (const float*)d_in[4];   // (3H, H)
    const float* b_ih = (const float*)d_in[5];   // (3H)
    const float* b_hh = (const float*)d_in[6];   // (3H)
    (void)in_sizes; (void)n_in; (void)out_size; (void)ws_size;

    float* ws = (float*)d_ws;
    float* ws_scores  = ws;                                   // B*S
    float* ws_ctxpart = ws_scores  + B_SZ * S_LEN;            // B*NCHUNK*H
    float* ws_mlpart  = ws_ctxpart + B_SZ * NCHUNK * H_SZ;    // B*NCHUNK*2
    float* ws_mlfin   = ws_mlpart  + B_SZ * NCHUNK * 2;       // B*2
    float* ws_context = ws_mlfin   + B_SZ * 2;                // B*H
    float* ws_gi      = ws_context + B_SZ * H_SZ;             // B*3H
    float* ws_gh      = ws_gi      + B_SZ * N3H;              // B*3H

    float* out      = (float*)d_out;
    float* out_attn = out + 2 * B_SZ * H_SZ;                  // (B,1,S)

    attn_partial_kernel<<<B_SZ * NCHUNK, 256, 0, stream>>>(
        lh, enc, ws_scores, ws_ctxpart, ws_mlpart);

    attn_combine_kernel<<<B_SZ, 256, 0, stream>>>(
        ws_ctxpart, ws_mlpart, ws_context, ws_mlfin);

    attn_weights_kernel<<<(B_SZ * S_LEN) / 256, 256, 0, stream>>>(
        ws_scores, ws_mlfin, out_attn);

    gru_gemm_kernel<<<N3H / 32, 256, 0, stream>>>(
        x, ws_context, lh, W_ih, W_hh, b_ih, b_hh, ws_gi, ws_gh);

    gru_gate_kernel<<<(B_SZ * H_SZ) / 256, 256, 0, stream>>>(
        ws_gi, ws_gh, lh, out);
}